// rinter_area_compute_23639499997217
// MI455X (gfx1250) — compile-verified
//
#include <hip/hip_runtime.h>
#include <hip/hip_bf16.h>

// Streaming, HBM-bound kernel for triangle-fan area with count mask.
// Per row: 4x GLOBAL_LOAD_B128 (non-temporal) + 1x b32 load + ~40 VALU + 1x b32 store.
// Roofline: 288 MB @ 23.3 TB/s => ~12.4 us floor; FLOPs negligible (0.24 GFLOP).

typedef float v4f __attribute__((ext_vector_type(4)));

__global__ __launch_bounds__(256) void rinter_area_kernel(
    const v4f* __restrict__ pts,      // N rows x 4 float4 (16 floats per row)
    const int* __restrict__ n_inter,  // N
    float*     __restrict__ out,      // N
    int n)
{
    int i = blockIdx.x * blockDim.x + threadIdx.x;
    if (i >= n) return;

    const v4f* row = pts + (size_t)i * 4;
    // Stream reads: data is touched exactly once and exceeds L2 (256MB > 192MB)
    v4f v0 = __builtin_nontemporal_load(row + 0);
    v4f v1 = __builtin_nontemporal_load(row + 1);
    v4f v2 = __builtin_nontemporal_load(row + 2);
    v4f v3 = __builtin_nontemporal_load(row + 3);
    int k  = __builtin_nontemporal_load(n_inter + i) - 2;   // triangles valid if t < k

    // 8 points (x,y)
    float px[8] = {v0.x, v0.z, v1.x, v1.z, v2.x, v2.z, v3.x, v3.z};
    float py[8] = {v0.y, v0.w, v1.y, v1.w, v2.y, v2.w, v3.y, v3.w};

    const float p1x = px[0];
    const float p1y = py[0];

    float acc = 0.0f;
#pragma unroll
    for (int t = 0; t < 6; ++t) {
        float p2x = px[t + 1], p2y = py[t + 1];
        float p3x = px[t + 2], p3y = py[t + 2];
        float cr = (p1x - p3x) * (p2y - p3y) - (p1y - p3y) * (p2x - p3x);
        float area = __builtin_fabsf(cr) * 0.5f;   // abs folds into src modifier
        acc += (t < k) ? area : 0.0f;              // v_cmp + v_cndmask
    }

    __builtin_nontemporal_store(acc, out + i);
}

extern "C" void kernel_launch(void* const* d_in, const int* in_sizes, int n_in,
                              void* d_out, int out_size, void* d_ws, size_t ws_size,
                              hipStream_t stream) {
    const v4f* pts     = (const v4f*)d_in[0];   // float32 (N,16) -> N x 4 float4
    const int* n_inter = (const int*)d_in[1];   // int32 (N,)
    float*     out     = (float*)d_out;         // float32 (N,)

    const int n = in_sizes[1];                  // N rows
    const int threads = 256;                    // 8 waves (wave32) per block
    const int blocks  = (n + threads - 1) / threads;

    rinter_area_kernel<<<blocks, threads, 0, stream>>>(pts, n_inter, out, n);
}